// KLDivergenceModule_21423296872469
// MI455X (gfx1250) — compile-verified
//
#include <hip/hip_runtime.h>

// ---------------------------------------------------------------------------
// Binary KL divergence, summed:  sum_i p*ln(p/q) + (1-p)*ln((1-p)/(1-q))
// Memory-bound streaming reduction. All math in log2 domain (v_log_f32),
// scaled by ln(2) once per block. Cross-lane reduction done by
// V_WMMA_F32_16X16X4_F32 with an all-ones B matrix (D = A x 1 + C), keeping
// full f32 precision while exercising the CDNA5 matrix pipe.
// ---------------------------------------------------------------------------

typedef __attribute__((ext_vector_type(2))) float v2f;
typedef __attribute__((ext_vector_type(4))) float v4f;
typedef __attribute__((ext_vector_type(8))) float v8f;

#define KL_NBLOCKS  4096
#define KL_NTHREADS 256
#define KL_NWAVES   (KL_NTHREADS / 32)
#define LN2F        0.69314718055994530942f

// Per-element KL term in log2 domain: p*(lg p - lg q) + (1-p)*(lg(1-p) - lg(1-q))
__device__ __forceinline__ float kl2_term(float p, float q) {
    float lp  = __builtin_amdgcn_logf(p);          // v_log_f32 (log2)
    float lq  = __builtin_amdgcn_logf(q);
    float l1p = __builtin_amdgcn_logf(1.0f - p);
    float l1q = __builtin_amdgcn_logf(1.0f - q);
    return p * (lp - lq) + (1.0f - p) * (l1p - l1q);
}

__global__ __launch_bounds__(KL_NTHREADS)
void kl_partial_kernel(const float* __restrict__ pin,
                       const float* __restrict__ qin,
                       float* __restrict__ partial,
                       unsigned long long n4) {
    const v4f* __restrict__ p4 = (const v4f*)pin;
    const v4f* __restrict__ q4 = (const v4f*)qin;

    const unsigned long long tid    = (unsigned long long)blockIdx.x * blockDim.x + threadIdx.x;
    const unsigned long long stride = (unsigned long long)gridDim.x * blockDim.x;
    const unsigned long long iters  = (n4 + stride - 1) / stride;   // wave-uniform trip count

    v8f acc = {0.f, 0.f, 0.f, 0.f, 0.f, 0.f, 0.f, 0.f};
    v2f ones; ones.x = 1.0f; ones.y = 1.0f;                         // B = 4x16 all-ones

    for (unsigned long long j = 0; j < iters; ++j) {
        unsigned long long i = tid + j * stride;
        v2f a; a.x = 0.0f; a.y = 0.0f;                              // zero for inactive lanes
        if (i < n4) {
            v4f p = __builtin_nontemporal_load(p4 + i);             // streamed once: NT hint
            v4f q = __builtin_nontemporal_load(q4 + i);
            float k0 = kl2_term(p.x, q.x);
            float k1 = kl2_term(p.y, q.y);
            float k2 = kl2_term(p.z, q.z);
            float k3 = kl2_term(p.w, q.w);
            a.x = k0 + k2;
            a.y = k1 + k3;
        }
        // EXEC is all-1s here (re-converged). D = A x ones + C: sums the wave's
        // 64 f32 terms into the f32 accumulator in one matrix op.
        acc = __builtin_amdgcn_wmma_f32_16x16x4_f32(
            /*neg_a=*/false, a, /*neg_b=*/false, ones,
            /*c_mod=*/(short)0, acc, /*reuse_a=*/false, /*reuse_b=*/false);
    }

    // Lane-local sum of the 8 D VGPRs. Layout: lanes 0-15 hold rows 0-7,
    // lanes 16-31 hold rows 8-15 (every column n is the identical row-sum
    // vector since B was all ones).
    float s = acc[0] + acc[1] + acc[2] + acc[3] + acc[4] + acc[5] + acc[6] + acc[7];
    float wave_total = __shfl(s, 0, 32) + __shfl(s, 16, 32);

    __shared__ float lds[KL_NWAVES];
    const unsigned int lane = threadIdx.x & 31u;
    const unsigned int wid  = threadIdx.x >> 5;
    if (lane == 0) lds[wid] = wave_total;
    __syncthreads();
    if (threadIdx.x == 0) {
        float b = 0.0f;
#pragma unroll
        for (int w = 0; w < KL_NWAVES; ++w) b += lds[w];
        partial[blockIdx.x] = b * LN2F;                             // back to natural log
    }
}

__global__ __launch_bounds__(KL_NTHREADS)
void kl_final_kernel(const float* __restrict__ partial, int nparts,
                     const float* __restrict__ pin, const float* __restrict__ qin,
                     unsigned long long tail_begin, unsigned long long n,
                     float* __restrict__ out) {
    __shared__ float lds[KL_NTHREADS];
    float s = 0.0f;
    for (int i = threadIdx.x; i < nparts; i += KL_NTHREADS) s += partial[i];
    lds[threadIdx.x] = s;
    __syncthreads();
    for (int off = KL_NTHREADS / 2; off > 0; off >>= 1) {
        if ((int)threadIdx.x < off) lds[threadIdx.x] += lds[threadIdx.x + off];
        __syncthreads();
    }
    if (threadIdx.x == 0) {
        float t = lds[0];
        // Scalar tail (n not divisible by 4): deterministic, tiny or empty.
        for (unsigned long long i = tail_begin; i < n; ++i)
            t += kl2_term(pin[i], qin[i]) * LN2F;
        out[0] = t;
    }
}

extern "C" void kernel_launch(void* const* d_in, const int* in_sizes, int n_in,
                              void* d_out, int out_size, void* d_ws, size_t ws_size,
                              hipStream_t stream) {
    (void)n_in; (void)out_size; (void)ws_size;
    const float* p = (const float*)d_in[0];
    const float* q = (const float*)d_in[1];
    float* out     = (float*)d_out;
    float* partial = (float*)d_ws;                    // KL_NBLOCKS floats of scratch

    const unsigned long long n  = (unsigned long long)in_sizes[0];
    const unsigned long long n4 = n >> 2;             // float4 chunks

    kl_partial_kernel<<<KL_NBLOCKS, KL_NTHREADS, 0, stream>>>(p, q, partial, n4);
    kl_final_kernel<<<1, KL_NTHREADS, 0, stream>>>(partial, KL_NBLOCKS,
                                                   p, q, n4 << 2, n, out);
}